// ImprovedHashEncoding_13967233646664
// MI455X (gfx1250) — compile-verified
//
#include <hip/hip_runtime.h>
#include <math.h>

#define N_LEVELS   16
#define N_FEAT     2
#define TABLE_SIZE (1u << 19)
#define TABLE_MASK (TABLE_SIZE - 1u)

#define P0 73856093u
#define P1 19349663u
#define P2 83492791u

struct ResArg { int res[N_LEVELS]; };

// 16 threads per point: lane handles (point n, level l). 256-thread block = 16 points.
// wave32: one wave = 2 points x 16 levels.
__global__ __launch_bounds__(256)
void hashenc_kernel(const float* __restrict__ x,
                    const float* __restrict__ tables,
                    float* __restrict__ out,
                    ResArg ra, int npoints) {
    // 16 points * 3 floats staged per block.
    __shared__ __align__(16) float scoord[48];

    const int tid   = threadIdx.x;
    const int gid0  = blockIdx.x * 256;
    const int pbase = gid0 >> 4;          // first point of this block

    if (pbase + 16 <= npoints) {
        // Full block: async-DMA 192B of coordinates into LDS (CDNA5 async-to-LDS path).
        if (tid < 12) {
            const int fidx = pbase * 3 + tid * 4;      // float index into x (16B aligned)
            // Low 32 bits of a generic pointer to LDS == wave-relative LDS byte offset.
            const unsigned lds_off =
                (unsigned)(unsigned long long)(uintptr_t)(&scoord[tid * 4]);
            const unsigned long long gaddr =
                (unsigned long long)(uintptr_t)(x + fidx);
            asm volatile("global_load_async_to_lds_b128 %0, %1, off"
                         :: "v"(lds_off), "v"(gaddr)
                         : "memory");
            // Warm L2 for coordinates ~128 blocks ahead (speculative stream prefetch).
            if (fidx + 6144 + 4 <= npoints * 3)
                __builtin_prefetch(x + fidx + 6144, 0, 3);
        }
        // Only wave 0 has ASYNCcnt != 0; harmless no-op for the others.
        asm volatile("s_wait_asynccnt 0x0" ::: "memory");
    } else {
        // Tail block: guarded per-element copy (uniform branch per block).
        if (tid < 48) {
            const int fi = pbase * 3 + tid;
            scoord[tid] = (fi < npoints * 3) ? x[fi] : 0.0f;
        }
    }
    __syncthreads();

    const int n = (gid0 + tid) >> 4;      // point index
    const int l = tid & 15;               // level index
    if (n >= npoints) return;
    const int p = tid >> 4;               // point index within block

    // Broadcast LDS reads (all 16 lanes of a point hit the same address).
    float px = scoord[p * 3 + 0];
    float py = scoord[p * 3 + 1];
    float pz = scoord[p * 3 + 2];
    px = fminf(fmaxf(px, 0.0f), 1.0f);    // jnp.clip(x, 0, 1)
    py = fminf(fmaxf(py, 0.0f), 1.0f);
    pz = fminf(fmaxf(pz, 0.0f), 1.0f);

    const int   res  = ra.res[l];
    const int   maxc = res - 1;
    const float rf   = (float)maxc;

    // scaled = x * (res - 1)
    const float sx = px * rf, sy = py * rf, sz = pz * rf;

    // floor corner, clipped to [0, maxc]
    int cfx = (int)floorf(sx); cfx = min(max(cfx, 0), maxc);
    int cfy = (int)floorf(sy); cfy = min(max(cfy, 0), maxc);
    int cfz = (int)floorf(sz); cfz = min(max(cfz, 0), maxc);

    // trilinear weights (w = scaled - float(cf))
    const float wx = sx - (float)cfx, omx = 1.0f - wx;
    const float wy = sy - (float)cfy, omy = 1.0f - wy;
    const float wz = sz - (float)cfz, omz = 1.0f - wz;

    // Hash is linear in corner offsets mod 2^32: base + delta(P or 0 if clamped).
    // Power-of-two mask at the end keeps this exact vs. the reference.
    const unsigned base = (unsigned)cfx * P0 + (unsigned)cfy * P1 + (unsigned)cfz * P2;
    const unsigned dX = (cfx < maxc) ? P0 : 0u;
    const unsigned dY = (cfy < maxc) ? P1 : 0u;
    const unsigned dZ = (cfz < maxc) ? P2 : 0u;

    const float2* __restrict__ tb =
        (const float2*)(tables + (size_t)l * (size_t)(TABLE_SIZE * N_FEAT));

    float acc0 = 0.0f, acc1 = 0.0f;

    // Reference corner enumeration: corner offsets (x,y,z) = (bit2, bit1, bit0),
    // but weight factors use (w_x <- bit0, w_y <- bit1, w_z <- bit2). Reproduced exactly,
    // including the i = 0..7 accumulation order.
#pragma unroll
    for (int i = 0; i < 8; ++i) {
        const int ox = (i >> 2) & 1;   // x corner offset
        const int oy = (i >> 1) & 1;   // y corner offset
        const int oz = i & 1;          // z corner offset

        unsigned h = base + (ox ? dX : 0u) + (oy ? dY : 0u) + (oz ? dZ : 0u);
        h &= TABLE_MASK;

        const float fwx = (i & 1)        ? wx : omx;  // mismatched pairing, per reference
        const float fwy = ((i >> 1) & 1) ? wy : omy;
        const float fwz = ((i >> 2) & 1) ? wz : omz;
        const float cw = (fwx * fwy) * fwz;

        const float2 f = tb[h];          // global_load_b64 gather (L2-resident table)
        acc0 += f.x * cw;
        acc1 += f.y * cw;
    }

    // out[n, l*2 + {0,1}] — consecutive lanes write consecutive float2: coalesced b64.
    float2* __restrict__ o = (float2*)(out + (size_t)n * (N_LEVELS * N_FEAT));
    o[l] = make_float2(acc0, acc1);
}

extern "C" void kernel_launch(void* const* d_in, const int* in_sizes, int n_in,
                              void* d_out, int out_size, void* d_ws, size_t ws_size,
                              hipStream_t stream) {
    const float* x      = (const float*)d_in[0];   // [N,3] f32
    const float* tables = (const float*)d_in[1];   // [16, 2^19, 2] f32
    float* out          = (float*)d_out;           // [N, 32] f32

    const int npoints = in_sizes[0] / 3;

    // Replicate Python's host-double RES computation exactly (same libm on host):
    //   b = exp((log(512) - log(16)) / 15);  RES[l] = min(int(16 * b**l), 512)
    ResArg ra;
    const double b = exp((log(512.0) - log(16.0)) / 15.0);
    for (int l = 0; l < N_LEVELS; ++l) {
        const double r = 16.0 * pow(b, (double)l);
        int ri = (int)r;                 // truncation toward zero, like Python int()
        if (ri > 512) ri = 512;
        ra.res[l] = ri;
    }

    const long long total = (long long)npoints * 16;
    const int block = 256;
    const int grid = (int)((total + block - 1) / block);

    hashenc_kernel<<<grid, block, 0, stream>>>(x, tables, out, ra, npoints);
}